// ConceptGraph_89970974916666
// MI455X (gfx1250) — compile-verified
//
#include <hip/hip_runtime.h>

typedef __attribute__((ext_vector_type(2))) float v2f;
typedef __attribute__((ext_vector_type(8))) float v8f;

#define D_MODEL 512
#define N_CONCEPTS 32
#define BT_TOKENS 65536
#define TOK_PER_WAVE 16
#define WAVES_PER_BLOCK 8
#define TOK_PER_BLOCK (TOK_PER_WAVE * WAVES_PER_BLOCK)

// ws layout: ws[0] = loss accumulator (float), ws[1..32] = counts (int)
__global__ void vq_zero_ws(int* ws) {
    int t = threadIdx.x;
    if (t < 33) ws[t] = 0;
}

__global__ __launch_bounds__(256) void vq_main(const float* __restrict__ x,
                                               const float* __restrict__ cb,
                                               float* __restrict__ out,
                                               float* __restrict__ loss_ws,
                                               int* __restrict__ counts_ws) {
    __shared__ float c2[N_CONCEPTS];
    __shared__ float s_mval[WAVES_PER_BLOCK][TOK_PER_WAVE];
    __shared__ int   s_midx[WAVES_PER_BLOCK][TOK_PER_WAVE];

    const int tid  = threadIdx.x;
    const int lane = tid & 31;
    const int wave = tid >> 5;

    // ---- cooperative |c|^2 : 256 threads x 64 contiguous floats (one code per 8 threads)
    if (tid < N_CONCEPTS) c2[tid] = 0.f;
    __syncthreads();
    {
        const float4* cb4 = (const float4*)cb;
        float p = 0.f;
        const int base = tid * 16; // 16 float4 = 64 floats, all within code tid/8
        #pragma unroll
        for (int i = 0; i < 16; ++i) {
            float4 v = cb4[base + i];
            p = fmaf(v.x, v.x, p); p = fmaf(v.y, v.y, p);
            p = fmaf(v.z, v.z, p); p = fmaf(v.w, v.w, p);
        }
        atomicAdd(&c2[tid >> 3], p);
    }
    __syncthreads();

    const int tokBase = blockIdx.x * TOK_PER_BLOCK + wave * TOK_PER_WAVE;
    const int m  = lane & 15;        // token row (A) / code column (B) within tile
    const int kk = (lane >> 4) * 2;  // K sub-offset per half-wave

    const float* xrow  = x  + (size_t)(tokBase + m) * D_MODEL + kk;
    const float* b0row = cb + (size_t)m * D_MODEL + kk;           // codes 0..15
    const float* b1row = cb + (size_t)(m + 16) * D_MODEL + kk;    // codes 16..31

    v8f acc0 = {};   // x . c^T for codes 0..15   (16x16 f32 tile)
    v8f acc1 = {};   // x . c^T for codes 16..31
    float x2p = 0.f; // per-lane partial of |x|^2 (this lane's K slices)

    #pragma unroll 4
    for (int k0 = 0; k0 < D_MODEL; k0 += 4) {
        float2 a  = *(const float2*)(xrow  + k0);
        float2 b0 = *(const float2*)(b0row + k0);
        float2 b1 = *(const float2*)(b1row + k0);
        x2p = fmaf(a.x, a.x, x2p);
        x2p = fmaf(a.y, a.y, x2p);
        v2f av;  av[0]  = a.x;  av[1]  = a.y;
        v2f bv0; bv0[0] = b0.x; bv0[1] = b0.y;
        v2f bv1; bv1[0] = b1.x; bv1[1] = b1.y;
        acc0 = __builtin_amdgcn_wmma_f32_16x16x4_f32(false, av, false, bv0,
                                                     (short)0, acc0, false, false);
        acc1 = __builtin_amdgcn_wmma_f32_16x16x4_f32(false, av, false, bv1,
                                                     (short)0, acc1, false, false);
    }
    // full |x|^2 for token m lives on lanes m and m+16 after xor-16 combine
    float x2 = x2p + __shfl_xor(x2p, 16);

    const float c2a = c2[m];
    const float c2b = c2[m + 16];

    // ---- per-token argmin over 32 codes.
    // C-tile layout: VGPR r -> token r (lanes 0..15, code = lane) and token r+8
    // (lanes 16..31, code = lane-16). x^2 is constant per token, so argmin of
    // (c^2 - 2 x.c) suffices; add x^2 afterwards for the loss value.
    #pragma unroll
    for (int r = 0; r < 8; ++r) {
        float v0 = fmaf(-2.f, acc0[r], c2a);
        float v1 = fmaf(-2.f, acc1[r], c2b);
        float mv; int mi;
        if (v1 < v0) { mv = v1; mi = m + 16; } else { mv = v0; mi = m; }
        #pragma unroll
        for (int s = 1; s <= 8; s <<= 1) {   // butterfly within 16-lane groups
            float ov = __shfl_xor(mv, s);
            int   oi = __shfl_xor(mi, s);
            if (ov < mv || (ov == mv && oi < mi)) { mv = ov; mi = oi; } // first-min tie-break
        }
        if (m == 0) {
            int tok = r + (lane >> 4) * 8;
            s_mval[wave][tok] = mv;
            s_midx[wave][tok] = mi;
        }
    }
    __syncthreads();

    // ---- loss + histogram: lanes 0..15 own tokens 0..15 of this wave
    float dist = 0.f;
    if (lane < 16) {
        dist = x2 + s_mval[wave][lane];        // = |q - x|^2 for this token
        atomicAdd(&counts_ws[s_midx[wave][lane]], 1);
    }
    #pragma unroll
    for (int s = 1; s <= 8; s <<= 1) dist += __shfl_xor(dist, s);
    if (lane == 0) atomicAdd(loss_ws, dist);

    // ---- straight-through output: out = x + (q - x), q = codebook row (L2-resident)
    #pragma unroll 4
    for (int t = 0; t < TOK_PER_WAVE; ++t) {
        const int mi = s_midx[wave][t];
        const float4* qsrc = (const float4*)(cb + (size_t)mi * D_MODEL);
        const float4* xsrc = (const float4*)(x + (size_t)(tokBase + t) * D_MODEL);
        float4*       dst  = (float4*)(out + (size_t)(tokBase + t) * D_MODEL);
        #pragma unroll
        for (int j = 0; j < 4; ++j) {         // 32 lanes x float4 x 4 = 512 floats
            int e = lane + j * 32;
            float4 q  = qsrc[e];
            float4 xv = xsrc[e];
            float4 o;
            o.x = xv.x + (q.x - xv.x);
            o.y = xv.y + (q.y - xv.y);
            o.z = xv.z + (q.z - xv.z);
            o.w = xv.w + (q.w - xv.w);
            dst[e] = o;
        }
    }
}

__global__ void vq_finalize(const float* __restrict__ loss_ws,
                            const int* __restrict__ counts_ws,
                            float* __restrict__ out_tail) {
    int t = threadIdx.x;
    if (t == 0) {
        out_tail[0] = 1.25f * loss_ws[0] / (float)((size_t)BT_TOKENS * D_MODEL);
    }
    if (t < N_CONCEPTS) {
        out_tail[1 + t] = (float)counts_ws[t];
    }
}

extern "C" void kernel_launch(void* const* d_in, const int* in_sizes, int n_in,
                              void* d_out, int out_size, void* d_ws, size_t ws_size,
                              hipStream_t stream) {
    const float* x  = (const float*)d_in[0];   // [8,8192,512] f32
    const float* cb = (const float*)d_in[1];   // [32,512] f32
    float* out = (float*)d_out;                // [quantized | loss | counts]
    float* loss_ws   = (float*)d_ws;
    int*   counts_ws = (int*)d_ws + 1;

    vq_zero_ws<<<1, 64, 0, stream>>>((int*)d_ws);
    vq_main<<<BT_TOKENS / TOK_PER_BLOCK, 256, 0, stream>>>(x, cb, out, loss_ws, counts_ws);
    vq_finalize<<<1, 64, 0, stream>>>(loss_ws, counts_ws,
                                      out + (size_t)BT_TOKENS * D_MODEL);
}